// RelGraphEmbedLayer_49460843381423
// MI455X (gfx1250) — compile-verified
//
#include <hip/hip_runtime.h>
#include <hip/hip_bf16.h>
#include <stdint.h>

typedef __attribute__((ext_vector_type(2))) float v2f;
typedef __attribute__((ext_vector_type(8))) float v8f;

// ---------------------------------------------------------------------------
// GEMM: out[loc0[m], :] = feat0[m, :] (512) @ W0 (512x128)
// Block: 256 threads (8 waves) -> 128(M) x 128(N) tile.
// Each wave: one 16-row M tile x 8 N tiles via V_WMMA_F32_16X16X4_F32.
// Global->LDS staging uses async-to-LDS DMA (ASYNCcnt) with double buffering.
// ---------------------------------------------------------------------------
#define TILE_M    128
#define TILE_K    32
#define AS_STRIDE 36                 // floats; 144 B rows -> 16B-aligned b128 dst
#define BT_STRIDE 34                 // floats; 136 B rows -> 8B-aligned b64 reads
#define A_FLOATS  (TILE_M * AS_STRIDE)   // 4608
#define B_FLOATS  (128 * BT_STRIDE)      // 4352

__global__ __launch_bounds__(256)
void gemm_scatter_kernel(const float* __restrict__ feat0,
                         const float* __restrict__ W0,
                         const int* __restrict__ loc0,
                         float* __restrict__ out)
{
    __shared__ float lds[2 * A_FLOATS + 2 * B_FLOATS];   // 71680 B

    const int tid    = threadIdx.x;
    const int lane   = tid & 31;
    const int wave   = tid >> 5;          // 0..7
    const int blockM = blockIdx.x * TILE_M;
    const int mtile  = wave * 16;
    const int l15    = lane & 15;
    const int kofs   = (lane >> 4) * 2;   // A/B operand: lanes 0-15 K={0,1}, 16-31 K={2,3}

    // LDS byte offset of the shared block (low 32 bits of flat LDS address).
    const uint32_t lds0  = (uint32_t)(uintptr_t)&lds[0];
    const uint64_t featp = (uint64_t)(uintptr_t)feat0;
    const uint64_t w0p   = (uint64_t)(uintptr_t)W0;

    // Issue one K-chunk of async global->LDS DMA (20 async ops per wave):
    //   A tile : 128x32 f32, 4 x b128 per thread (coalesced, padded rows)
    //   B chunk: W0[kk+k][n] -> Bt[n][k] transpose, 16 x b32 per thread
    auto issue_chunk = [&](int kk, int buf) {
        const uint32_t abase = lds0 + (uint32_t)(buf * A_FLOATS) * 4u;
        const uint32_t bbase = lds0 + (uint32_t)(2 * A_FLOATS + buf * B_FLOATS) * 4u;
#pragma unroll
        for (int i = 0; i < 4; ++i) {
            int linear = tid + i * 256;              // float4 index in tile
            int row    = linear >> 3;                // 8 float4 per 32-col row
            int col    = (linear & 7) << 2;
            uint32_t goff = (uint32_t)(((blockM + row) * 512 + kk + col) * 4);
            uint32_t loff = abase + (uint32_t)(row * AS_STRIDE + col) * 4u;
            asm volatile("global_load_async_to_lds_b128 %0, %1, %2 offset:0"
                         :: "v"(loff), "v"(goff), "s"(featp) : "memory");
        }
#pragma unroll
        for (int i = 0; i < 16; ++i) {
            int linear = tid + i * 256;
            int n = linear & 127;
            int k = linear >> 7;
            uint32_t goff = (uint32_t)(((kk + k) * 128 + n) * 4);
            uint32_t loff = bbase + (uint32_t)(n * BT_STRIDE + k) * 4u;
            asm volatile("global_load_async_to_lds_b32 %0, %1, %2 offset:0"
                         :: "v"(loff), "v"(goff), "s"(w0p) : "memory");
        }
    };

    v8f acc[8];
#pragma unroll
    for (int i = 0; i < 8; ++i) acc[i] = v8f{};

    issue_chunk(0, 0);                    // prologue: chunk 0 -> buffer 0

    for (int c = 0; c < 512 / TILE_K; ++c) {
        if (c + 1 < 512 / TILE_K) {
            // prefetch next chunk into the other buffer, then wait for the
            // oldest 20 async ops (= this chunk) while next stays in flight
            issue_chunk((c + 1) * TILE_K, (c + 1) & 1);
            asm volatile("s_wait_asynccnt 20" ::: "memory");
        } else {
            asm volatile("s_wait_asynccnt 0" ::: "memory");
        }
        __syncthreads();                  // chunk c visible to all waves

        const int ab = (c & 1) * A_FLOATS;
        const int bb = 2 * A_FLOATS + (c & 1) * B_FLOATS;
        const int arow_off = ab + (mtile + l15) * AS_STRIDE + kofs;
#pragma unroll
        for (int k0 = 0; k0 < TILE_K; k0 += 4) {
            v2f a = *(const v2f*)&lds[arow_off + k0];
            v2f b[8];
#pragma unroll
            for (int nt = 0; nt < 8; ++nt)
                b[nt] = *(const v2f*)&lds[bb + (nt * 16 + l15) * BT_STRIDE + k0 + kofs];
#pragma unroll
            for (int nt = 0; nt < 8; ++nt)
                acc[nt] = __builtin_amdgcn_wmma_f32_16x16x4_f32(
                    /*neg_a=*/false, a, /*neg_b=*/false, b[nt],
                    /*c_mod=*/(short)0, acc[nt],
                    /*reuse_a=*/false, /*reuse_b=*/false);
        }
        __syncthreads();                  // safe to overwrite buffer (c&1)
    }

    // --- Epilogue: scatter rows through loc0 ---
    // C/D layout: VGPR r -> lanes 0-15: (M=mtile+r, N=lane), lanes 16-31: (M=mtile+8+r)
    int rowids = loc0[blockM + mtile + l15];     // lane L<16 holds row id of M-row L
#pragma unroll
    for (int r = 0; r < 8; ++r) {
        int src  = r + ((lane >> 4) << 3);       // r (lo half) / r+8 (hi half)
        int orow = __shfl(rowids, src, 32);
#pragma unroll
        for (int nt = 0; nt < 8; ++nt) {
            out[(size_t)orow * 128 + nt * 16 + l15] = acc[nt][r];
        }
    }
}

// ---------------------------------------------------------------------------
// Gather: out[loc, :] = emb_weight[node_ids[loc], :] for loc in loc1 ++ loc2
// One wave32 per 128-float row; one float4 per lane (512 B / row).
// ---------------------------------------------------------------------------
__global__ __launch_bounds__(256)
void gather_embed_kernel(const int* __restrict__ node_ids,
                         const int* __restrict__ loc1,
                         const int* __restrict__ loc2,
                         const float* __restrict__ emb,
                         float* __restrict__ out,
                         int n1, int n_total)
{
    int gt   = blockIdx.x * 256 + threadIdx.x;
    int row  = gt >> 5;
    int lane = gt & 31;
    if (row >= n_total) return;

    int loc = (row < n1) ? loc1[row] : loc2[row - n1];
    int nid = node_ids[loc];

    const float4* src = (const float4*)&emb[(size_t)nid * 128];
    float4*       dst = (float4*)&out[(size_t)loc * 128];
    dst[lane] = src[lane];
}

extern "C" void kernel_launch(void* const* d_in, const int* in_sizes, int n_in,
                              void* d_out, int out_size, void* d_ws, size_t ws_size,
                              hipStream_t stream)
{
    // setup_inputs order: node_ids, loc0, loc1, loc2, feat0, W0, emb_weight
    const int*   node_ids = (const int*)d_in[0];
    const int*   loc0     = (const int*)d_in[1];
    const int*   loc1     = (const int*)d_in[2];
    const int*   loc2     = (const int*)d_in[3];
    const float* feat0    = (const float*)d_in[4];
    const float* W0       = (const float*)d_in[5];
    const float* emb      = (const float*)d_in[6];
    float*       out      = (float*)d_out;

    const int n0 = in_sizes[1];              // 262144
    const int n1 = in_sizes[2];              // 393216
    const int n2 = in_sizes[3];              // 393216
    const int n_featureless = n1 + n2;       // 786432

    dim3 gemm_grid(n0 / TILE_M);
    gemm_scatter_kernel<<<gemm_grid, 256, 0, stream>>>(feat0, W0, loc0, out);

    dim3 gather_grid(((size_t)n_featureless * 32 + 255) / 256);
    gather_embed_kernel<<<gather_grid, 256, 0, stream>>>(node_ids, loc1, loc2, emb, out,
                                                         n1, n_featureless);
}